// GraphSAGE_15745350107889
// MI455X (gfx1250) — compile-verified
//
#include <hip/hip_runtime.h>

// GraphSAGE (2-layer, mean aggr) for MI455X / gfx1250.
// Edge scatter dominates (~3.3GB random traffic, L2-resident at 192MB);
// GEMMs are ~8.6 GFLOP total, so we keep exact fp32 via V_WMMA_F32_16X16X4_F32.
// A-tiles (agg + self, 16 rows x K) are staged once per block into LDS with a
// +4-float row pad (bank-conflict-free ds_load_b64 in the K loop), mean-scale
// fused into the staging store. Layer-2 weights zero-padded 40->48 rows so all
// inner-loop loads are unconditional.

#define N_NODES   100000
#define D_IN      128
#define D_HID     128
#define D_OUT     40
#define D_OUT_PAD 48

typedef __attribute__((ext_vector_type(2))) float v2f;
typedef __attribute__((ext_vector_type(8))) float v8f;

__global__ void fill_zero_f32(float* __restrict__ p, int n) {
  int i = blockIdx.x * blockDim.x + threadIdx.x;
  if (i < n) p[i] = 0.0f;
}

// Copy [srcRows, K] -> [dstRows, K], zero-filling rows >= srcRows.
__global__ void pad_rows_f32(const float* __restrict__ src, float* __restrict__ dst,
                             int srcRows, int dstRows, int K) {
  int i = blockIdx.x * blockDim.x + threadIdx.x;
  if (i >= dstRows * K) return;
  int r = i / K;
  dst[i] = (r < srcRows) ? src[i] : 0.0f;
}

// deg[dst] += 1 per edge
__global__ void degree_kernel(const int* __restrict__ dst,
                              float* __restrict__ deg, int E) {
  int e = blockIdx.x * blockDim.x + threadIdx.x;
  if (e < E) unsafeAtomicAdd(&deg[dst[e]], 1.0f);
}

// One wave (32 lanes) per edge; lane l owns features [4l, 4l+4).
// Row read is a coalesced global_load_b128; scatter is 4x global_atomic_add_f32.
__global__ void scatter_kernel(const float* __restrict__ X,
                               const int* __restrict__ src,
                               const int* __restrict__ dst,
                               float* __restrict__ agg, int E) {
  int wave = (blockIdx.x * blockDim.x + threadIdx.x) >> 5;
  int lane = threadIdx.x & 31;
  if (wave >= E) return;
  int s = src[wave];
  int d = dst[wave];
  const float4* xr = (const float4*)(X + (long)s * D_HID);
  float4 v = xr[lane];
  float* ar = agg + (long)d * D_HID + lane * 4;
  unsafeAtomicAdd(ar + 0, v.x);
  unsafeAtomicAdd(ar + 1, v.y);
  unsafeAtomicAdd(ar + 2, v.z);
  unsafeAtomicAdd(ar + 3, v.w);
}

// Fused SAGEConv GEMM: out = act( (agg/deg) @ Wl^T + self @ Wr^T + bias )
// Block = NTILES waves; wave w owns output column tile [16w, 16w+16).
// Stage: 16xK agg tile (pre-scaled by 1/deg) + 16xK self tile -> LDS.
// Compute: per k-step two ds_load_b64 (A), two global_load_b64 (B),
//          two V_WMMA_F32_16X16X4_F32 into one fp32 accumulator.
template <int K, int NTILES, int NSTORE, bool RELU>
__global__ void __launch_bounds__(256)
sage_gemm_wmma(const float* __restrict__ Aagg, const float* __restrict__ Aself,
               const float* __restrict__ deg,
               const float* __restrict__ Wl, const float* __restrict__ Wr,
               const float* __restrict__ bias,
               float* __restrict__ out) {
  constexpr int STRIDE = K + 4;           // LDS row stride (floats): bank-safe
  __shared__ float ldsA[16 * STRIDE];     // mean-scaled aggregate tile
  __shared__ float ldsS[16 * STRIDE];     // self tile

  const int tid  = threadIdx.x;
  const int row0 = blockIdx.x * 16;       // N_NODES = 6250 * 16 exactly

  // ---- cooperative stage (float4 units), scale fused into agg tile ----
  const float4* a4 = (const float4*)Aagg;
  const float4* s4 = (const float4*)Aself;
  for (int fi = tid; fi < 16 * (K / 4); fi += blockDim.x) {
    int row = fi / (K / 4);
    int c4  = fi % (K / 4);
    float sc = 1.0f / fmaxf(deg[row0 + row], 1.0f);
    float4 va = a4[(size_t)(row0 + row) * (K / 4) + c4];
    float4 vs = s4[(size_t)(row0 + row) * (K / 4) + c4];
    va.x *= sc; va.y *= sc; va.z *= sc; va.w *= sc;
    *(float4*)&ldsA[row * STRIDE + c4 * 4] = va;   // ds_store_b128
    *(float4*)&ldsS[row * STRIDE + c4 * 4] = vs;
  }
  __syncthreads();

  // ---- per-wave 16x16 WMMA tile ----
  const int lane = tid & 31;
  const int wv   = tid >> 5;              // < NTILES by launch config
  const int half = lane >> 4;             // 0: lanes 0-15, 1: lanes 16-31
  const int sub  = lane & 15;
  const int koff = half * 2;              // A/B VGPR k placement per ISA layout
  const int bcol = wv * 16 + sub;         // B column owned by this lane

  const float* aRow  = &ldsA[sub * STRIDE + koff];
  const float* sRow  = &ldsS[sub * STRIDE + koff];
  const float* wlRow = Wl + (size_t)bcol * K + koff;
  const float* wrRow = Wr + (size_t)bcol * K + koff;

  v8f acc = {};
#pragma unroll 4
  for (int k = 0; k < K; k += 4) {
    v2f aL = *(const v2f*)(aRow + k);     // ds_load_b64, conflict-free
    v2f aR = *(const v2f*)(sRow + k);
    v2f bL = *(const v2f*)(wlRow + k);    // global_load_b64, L2-resident
    v2f bR = *(const v2f*)(wrRow + k);
    acc = __builtin_amdgcn_wmma_f32_16x16x4_f32(
        false, aL, false, bL, (short)0, acc, false, false);
    acc = __builtin_amdgcn_wmma_f32_16x16x4_f32(
        false, aR, false, bR, (short)0, acc, false, false);
  }

  const float bval = bias[bcol];
  // C/D layout: VGPR v -> M = v + 8*half, N = sub (both lane halves)
  const int ocol = wv * 16 + sub;
#pragma unroll
  for (int v = 0; v < 8; ++v) {
    int orow = row0 + v + half * 8;
    float val = acc[v] + bval;
    if (RELU) val = fmaxf(val, 0.0f);
    if (ocol < NSTORE) out[(size_t)orow * NSTORE + ocol] = val;
  }
}

extern "C" void kernel_launch(void* const* d_in, const int* in_sizes, int n_in,
                              void* d_out, int out_size, void* d_ws, size_t ws_size,
                              hipStream_t stream) {
  const float* x   = (const float*)d_in[0];
  const int*  edge = (const int*)d_in[1];   // [2, E] (src row then dst row)
  const float* W1l = (const float*)d_in[2];
  const float* b1  = (const float*)d_in[3];
  const float* W1r = (const float*)d_in[4];
  const float* W2l = (const float*)d_in[5];
  const float* b2  = (const float*)d_in[6];
  const float* W2r = (const float*)d_in[7];
  float* out = (float*)d_out;

  const int E = in_sizes[1] / 2;
  const int* src = edge;
  const int* dst = edge + E;

  // Workspace: deg[N] | agg[N*128] | h[N*128] | W2l_pad | W2r_pad | b2_pad
  float* deg  = (float*)d_ws;
  float* agg  = deg + N_NODES;
  float* h    = agg + (size_t)N_NODES * D_HID;
  float* W2lp = h + (size_t)N_NODES * D_HID;
  float* W2rp = W2lp + D_OUT_PAD * D_HID;
  float* b2p  = W2rp + D_OUT_PAD * D_HID;

  const int ZT = 256;

  // ---- Prep: pad layer-2 weights/bias so GEMM loads are unconditional ----
  {
    int n = D_OUT_PAD * D_HID;
    pad_rows_f32<<<(n + ZT - 1) / ZT, ZT, 0, stream>>>(W2l, W2lp, D_OUT, D_OUT_PAD, D_HID);
    pad_rows_f32<<<(n + ZT - 1) / ZT, ZT, 0, stream>>>(W2r, W2rp, D_OUT, D_OUT_PAD, D_HID);
    pad_rows_f32<<<1, D_OUT_PAD, 0, stream>>>(b2, b2p, D_OUT, D_OUT_PAD, 1);
  }

  // ---- Layer 1 ----
  {
    int n = N_NODES * (D_HID + 1);             // deg + agg are contiguous
    fill_zero_f32<<<(n + ZT - 1) / ZT, ZT, 0, stream>>>(deg, n);
  }
  degree_kernel<<<(E + ZT - 1) / ZT, ZT, 0, stream>>>(dst, deg, E);
  scatter_kernel<<<(E + 7) / 8, 256, 0, stream>>>(x, src, dst, agg, E);
  // 8 waves/block -> 8 column tiles (128 cols); 6250 blocks of 16 rows
  sage_gemm_wmma<D_IN, 8, D_HID, true><<<N_NODES / 16, 256, 0, stream>>>(
      agg, x, deg, W1l, W1r, b1, h);

  // ---- Layer 2 ----
  {
    int n = N_NODES * D_HID;
    fill_zero_f32<<<(n + ZT - 1) / ZT, ZT, 0, stream>>>(agg, n);
  }
  scatter_kernel<<<(E + 7) / 8, 256, 0, stream>>>(h, src, dst, agg, E);
  // 3 waves/block cover 48 padded cols; cols >= 40 dropped at store
  sage_gemm_wmma<D_HID, 3, D_OUT, false><<<N_NODES / 16, 96, 0, stream>>>(
      agg, h, deg, W2lp, W2rp, b2p, out);
}